// Clustering_73744588472727
// MI455X (gfx1250) — compile-verified
//
#include <hip/hip_runtime.h>
#include <hip/hip_bf16.h>

// ---- CDNA5 WMMA types ----
typedef __attribute__((ext_vector_type(2))) float v2f;
typedef __attribute__((ext_vector_type(8))) float v8f;

__device__ inline v8f wmma_f32_4(v2f a, v2f b, v8f c) {
  // D(16x16,f32) = A(16x4,f32) * B(4x16,f32) + C
  return __builtin_amdgcn_wmma_f32_16x16x4_f32(false, a, false, b, (short)0, c,
                                               false, false);
}

// CDNA5 async global->LDS copy (GVS/GV encoding, tracked by ASYNCcnt).
// vdst VGPR holds the per-lane LDS byte offset; vaddr pair holds the 64-bit
// global address. Generic pointers to LDS are {aperture_hi | offset32}, so
// truncation yields the LDS offset the instruction wants.
__device__ inline void async_b128_to_lds(void* lds, const float* g) {
  unsigned loff = (unsigned)(unsigned long long)lds;
  unsigned long long ga = (unsigned long long)g;
  asm volatile("global_load_async_to_lds_b128 %0, %1, off" ::"v"(loff),
               "v"(ga)
               : "memory");
}
__device__ inline void wait_async0() {
  asm volatile("s_wait_asynccnt 0x0" ::: "memory");
}

#define NB 8
#define NH 8
#define LSEQ 1024
#define DH 64
#define DM 512
#define NCL 5
#define HLD (NH * LSEQ * DH)  // 524288
#define TPAD 68               // padded tile row stride: 272B = 16B-aligned,
                              // spreads 16-row column reads over banks

// ============================================================
// Kernel Init: zero the two scalar accumulators (lp, ce)
// ============================================================
__global__ void kInit(float* acc) { acc[threadIdx.x] = 0.f; }

// ============================================================
// Kernel A: per (b, l2)-site: unfold-gather + conv1x3 (cin=512),
// cluster softmaxes, loss terms, 5x5 masked cluster attention.
// One wave32 per site. 8192 sites.
// ============================================================
__global__ __launch_bounds__(32) void kA_clusters(
    const float* __restrict__ K, const float* __restrict__ w_pk,
    const float* __restrict__ b_pk, const float* __restrict__ w_ck,
    const float* __restrict__ b_ckv, const float* __restrict__ w_cq,
    const float* __restrict__ b_cqv, float* __restrict__ mu_ws,
    float* __restrict__ q2_ws, float* __restrict__ acc_g) {
  const int site = blockIdx.x;  // 0..8191
  const int b = site >> 10;
  const int p = site & 1023;
  const int lane = threadIdx.x;

  float acc[NCL][NCL];  // [nc][q]
#pragma unroll
  for (int nc = 0; nc < NCL; ++nc)
#pragma unroll
    for (int q = 0; q < NCL; ++q) acc[nc][q] = 0.f;

  const int d_idx = p & 63;
  const int phi = p >> 6;

  for (int cin = lane; cin < DM; cin += 32) {
    const int hh = cin >> 6;
    const int lidx = ((cin & 63) << 4) + phi;
    const int base = (hh * LSEQ + lidx) * DH + d_idx;
    float v[NCL];
#pragma unroll
    for (int q = 0; q < NCL; ++q) {
      const int bq = b + q - 4;  // K_padded[b+q] = (b+q<4) ? 0 : K[b+q-4]
      v[q] = (bq >= 0) ? K[bq * HLD + base] : 0.f;
    }
    const float* wp = w_pk + cin * 3;  // w_pk[nc*1536 + cin*3 + t]
#pragma unroll
    for (int nc = 0; nc < NCL; ++nc) {
      const float w0 = wp[nc * 1536 + 0];
      const float w1 = wp[nc * 1536 + 1];
      const float w2 = wp[nc * 1536 + 2];
#pragma unroll
      for (int q = 0; q < NCL; ++q) {
        float s = v[q] * w1;  // tap dx=1
        if (q > 0) s += v[q - 1] * w0;
        if (q < 4) s += v[q + 1] * w2;
        acc[nc][q] += s;
      }
    }
  }
  // wave32 reduction of the 25 partial sums
#pragma unroll
  for (int nc = 0; nc < NCL; ++nc)
#pragma unroll
    for (int q = 0; q < NCL; ++q) {
      float s = acc[nc][q];
      for (int off = 16; off; off >>= 1) s += __shfl_xor(s, off, 32);
      acc[nc][q] = s;
    }

  if (lane == 0) {
    float row[NCL][NCL];  // ckp transposed: [q][nc]
#pragma unroll
    for (int q = 0; q < NCL; ++q)
#pragma unroll
      for (int nc = 0; nc < NCL; ++nc)
        row[q][nc] = fmaxf(acc[nc][q] + b_pk[nc], 0.f);

    float ck[NCL][NCL], cq[NCL][NCL];
#pragma unroll
    for (int q = 0; q < NCL; ++q) {
      float lk[NCL], lq[NCL];
      for (int j = 0; j < NCL; ++j) {
        float sk = b_ckv[j], sq = b_cqv[j];
        for (int i = 0; i < NCL; ++i) {
          sk += row[q][i] * w_ck[j * 5 + i];
          sq += row[q][i] * w_cq[j * 5 + i];
        }
        lk[j] = sk;
        lq[j] = sq;
      }
      float mk = lk[0], mq = lq[0];
      for (int j = 1; j < NCL; ++j) {
        mk = fmaxf(mk, lk[j]);
        mq = fmaxf(mq, lq[j]);
      }
      float sk = 0.f, sq = 0.f;
      for (int j = 0; j < NCL; ++j) {
        ck[q][j] = expf(lk[j] - mk);
        sk += ck[q][j];
        cq[q][j] = expf(lq[j] - mq);
        sq += cq[q][j];
      }
      const float ik = 1.f / sk, iq = 1.f / sq;
      for (int j = 0; j < NCL; ++j) {
        ck[q][j] *= ik;
        cq[q][j] *= iq;
      }
    }
    // loss pieces: mu, sigma (softplus of unbiased std), x, log_prob
    float lp_local = 0.f;
#pragma unroll
    for (int q = 0; q < NCL; ++q) {
      float mu = 0.f, x = 0.f;
      for (int j = 0; j < NCL; ++j) {
        mu += cq[q][j];
        x += ck[q][j];
      }
      mu *= 0.2f;
      x *= 0.2f;
      float var = 0.f;
      for (int j = 0; j < NCL; ++j) {
        const float dv = cq[q][j] - mu;
        var += dv * dv;
      }
      var *= 0.25f;  // ddof=1
      const float sd = sqrtf(var);
      const float sigma = log1pf(expf(sd));  // softplus
      const float z = (x - mu) / sigma;
      lp_local += -0.5f * z * z - logf(sigma) - 0.91893853320467274f;
      mu_ws[site * NCL + q] = mu;
    }
    atomicAdd(&acc_g[0], lp_local);

    // masked triangular cluster attention (scores/5, mask u>p)
    float q2v[NCL][NCL];  // [p_row][c]
#pragma unroll
    for (int pr = 0; pr < NCL; ++pr) {
      float sc[NCL];
      float mx = -1e30f;
      for (int uq = 0; uq < NCL; ++uq) {
        float s;
        if (uq > pr) {
          s = -1e9f;
        } else {
          s = 0.f;
          for (int c = 0; c < NCL; ++c) s += cq[pr][c] * ck[uq][c];
          s *= 0.2f;
        }
        sc[uq] = s;
        mx = fmaxf(mx, s);
      }
      float ss = 0.f;
      for (int uq = 0; uq < NCL; ++uq) {
        sc[uq] = expf(sc[uq] - mx);
        ss += sc[uq];
      }
      const float inv = 1.f / ss;
      for (int c = 0; c < NCL; ++c) {
        float a = 0.f;
        for (int uq = 0; uq < NCL; ++uq) a += sc[uq] * cq[uq][c];
        q2v[pr][c] = a * inv;
      }
    }
    // store transposed (b, NC, l2, u) for conv-back
#pragma unroll
    for (int c = 0; c < NCL; ++c)
#pragma unroll
      for (int pr = 0; pr < NCL; ++pr)
        q2_ws[((b * NCL + c) * LSEQ + p) * NCL + pr] = q2v[pr][c];
  }
}

// ============================================================
// Kernel B: CE term — log_softmax over the l2=1024 axis per (b,u)
// ============================================================
__global__ __launch_bounds__(256) void kB_ce(const float* __restrict__ mu_ws,
                                             float* __restrict__ acc_g) {
  const int b = blockIdx.x / NCL;
  const int q = blockIdx.x % NCL;
  const int t = threadIdx.x;
  __shared__ float red[256];
  float vloc[4];
  float mx = -1e30f;
#pragma unroll
  for (int i = 0; i < 4; ++i) {
    const int l = t + i * 256;
    vloc[i] = mu_ws[((b << 10) + l) * NCL + q];
    mx = fmaxf(mx, vloc[i]);
  }
  red[t] = mx;
  __syncthreads();
  for (int s = 128; s; s >>= 1) {
    if (t < s) red[t] = fmaxf(red[t], red[t + s]);
    __syncthreads();
  }
  mx = red[0];
  __syncthreads();
  float se = 0.f;
#pragma unroll
  for (int i = 0; i < 4; ++i) se += expf(vloc[i] - mx);
  red[t] = se;
  __syncthreads();
  for (int s = 128; s; s >>= 1) {
    if (t < s) red[t] += red[t + s];
    __syncthreads();
  }
  const float lse = mx + logf(red[0]);
  __syncthreads();
  float part = 0.f;
#pragma unroll
  for (int i = 0; i < 4; ++i) part += vloc[i] * (vloc[i] - lse);
  red[t] = part;
  __syncthreads();
  for (int s = 128; s; s >>= 1) {
    if (t < s) red[t] += red[t + s];
    __syncthreads();
  }
  if (t == 0) atomicAdd(&acc_g[1], red[0]);
}

// ============================================================
// Kernel C: conv-back + scrambled reshape + sum over cluster axis
// -> CC2[b,h,l,d]. One thread per output element (4.19M).
// ============================================================
__global__ __launch_bounds__(256) void kC_center(
    const float* __restrict__ q2_ws, const float* __restrict__ w_back,
    const float* __restrict__ b_back, float* __restrict__ CC2) {
  const int tid = blockIdx.x * 256 + threadIdx.x;  // (b,h2,l3,d3) flat
  const int b = tid >> 19;
  const int fl = tid & ((1 << 19) - 1);
  const int h2 = fl >> 16;
  const int l3d3 = fl & 65535;
  const float* q2b = q2_ws + b * (NCL * LSEQ * NCL);
  float sum = 0.f;
#pragma unroll
  for (int u3 = 0; u3 < NCL; ++u3) {
    const int f = h2 * 327680 + u3 * 65536 + l3d3;  // flat in (c,l2,q) per b
    const int c = f / 5120;
    const int rem = f - c * 5120;
    const int l2 = rem / 5;
    const int qd = rem - l2 * 5;
    float a = b_back[c];
    const float* wr = w_back + c * 15;  // (512,5,1,3)
#pragma unroll
    for (int nc = 0; nc < NCL; ++nc) {
      const float* q2n = q2b + (nc * LSEQ + l2) * NCL;
      const float w0 = wr[nc * 3 + 0];
      const float w1 = wr[nc * 3 + 1];
      const float w2 = wr[nc * 3 + 2];
      float s = q2n[qd] * w1;
      if (qd > 0) s += q2n[qd - 1] * w0;
      if (qd < 4) s += q2n[qd + 1] * w2;
      a += s;
    }
    sum += fmaxf(a, 0.f);
  }
  CC2[tid] = sum;
}

// ============================================================
// Kernel E: finalize scalar loss
// ============================================================
__global__ void kE_loss(const float* __restrict__ acc_g,
                        float* __restrict__ out) {
  out[NB * NH * LSEQ * DH] = -(acc_g[0] / 40960.f) - (acc_g[1] / 40.f);
}

// ============================================================
// Kernel D: flash attention  out = softmax(Q @ CC2^T) @ V
// All 8 waves of a block share one (b,h) and walk the same kt
// sequence, so CC2/V 16x64 tiles are staged in LDS once per block
// via double-buffered global_load_async_to_lds_b128 (ASYNCcnt).
// WMMA: V_WMMA_F32_16X16X4_F32, 4 independent accumulator chains.
// ============================================================
__global__ __launch_bounds__(256) void kD_attn(const float* __restrict__ Q,
                                               const float* __restrict__ V,
                                               const float* __restrict__ CC2,
                                               float* __restrict__ out) {
  __shared__ __align__(16) float sCC[2][16][TPAD];
  __shared__ __align__(16) float sV[2][16][TPAD];
  __shared__ __align__(16) float Pt[8][16][16];
  const int t = threadIdx.x;
  const int wv = t >> 5;
  const int lane = t & 31;
  const int lrow = lane & 15;  // N (or M for A-frags)
  const int hi = lane >> 4;    // half-wave select
  const int widx = blockIdx.x * 8 + wv;  // 0..4095 ; same bh for whole block
  const int bh = widx >> 6;
  const int q0 = (widx & 63) << 4;

  // cooperative tile loaders: thread -> (row, 4-float column chunk)
  const int crow = t >> 4;
  const int ccol = (t & 15) << 2;
  const float* CCbase = CC2 + (bh * LSEQ + crow) * DH + ccol;
  const float* Vbase = V + (bh * LSEQ + crow) * DH + ccol;

  // A-fragments of the Q tile (16x64), f32 16x4 layout:
  // lanes 0-15: K = k0,k0+1 (M=lane); lanes 16-31: K = k0+2,k0+3 (M=lane-16)
  const float* Qrow = Q + (bh * LSEQ + q0 + lrow) * DH;
  v2f qa[16];
#pragma unroll
  for (int kc = 0; kc < 16; ++kc)
    qa[kc] = *(const v2f*)(Qrow + kc * 4 + hi * 2);

  const v8f zero8 = {0.f, 0.f, 0.f, 0.f, 0.f, 0.f, 0.f, 0.f};
  float m[8], lsum[8];
#pragma unroll
  for (int r = 0; r < 8; ++r) {
    m[r] = -1e30f;
    lsum[r] = 0.f;
  }
  v8f o[4];
#pragma unroll
  for (int j = 0; j < 4; ++j) o[j] = zero8;

  // prefetch kt = 0 into buffer 0
  async_b128_to_lds(&sCC[0][crow][ccol], CCbase);
  async_b128_to_lds(&sV[0][crow][ccol], Vbase);

  for (int kt = 0; kt < 64; ++kt) {
    const int cur = kt & 1;
    wait_async0();     // my async copies for buffer `cur` are done
    __syncthreads();   // everyone's are done; prior readers finished
    if (kt + 1 < 64) {
      const int off = (kt + 1) * 16 * DH;
      async_b128_to_lds(&sCC[cur ^ 1][crow][ccol], CCbase + off);
      async_b128_to_lds(&sV[cur ^ 1][crow][ccol], Vbase + off);
    }

    // ---- S = Q * CC2^T : 4 independent WMMA chains over K ----
    v8f s0 = zero8, s1 = zero8, s2 = zero8, s3 = zero8;
#pragma unroll
    for (int kc = 0; kc < 4; ++kc) {
      const v2f b0 = *(const v2f*)&sCC[cur][lrow][(kc + 0) * 4 + 2 * hi];
      const v2f b1 = *(const v2f*)&sCC[cur][lrow][(kc + 4) * 4 + 2 * hi];
      const v2f b2 = *(const v2f*)&sCC[cur][lrow][(kc + 8) * 4 + 2 * hi];
      const v2f b3 = *(const v2f*)&sCC[cur][lrow][(kc + 12) * 4 + 2 * hi];
      s0 = wmma_f32_4(qa[kc + 0], b0, s0);
      s1 = wmma_f32_4(qa[kc + 4], b1, s1);
      s2 = wmma_f32_4(qa[kc + 8], b2, s2);
      s3 = wmma_f32_4(qa[kc + 12], b3, s3);
    }
    v8f s = (s0 + s1) + (s2 + s3);

    // ---- online softmax per row (M = r + 8*hi in 16-lane groups) ----
    float scale[8];
#pragma unroll
    for (int r = 0; r < 8; ++r) {
      float v = s[r];
      v = fmaxf(v, __shfl_xor(v, 1, 32));
      v = fmaxf(v, __shfl_xor(v, 2, 32));
      v = fmaxf(v, __shfl_xor(v, 4, 32));
      v = fmaxf(v, __shfl_xor(v, 8, 32));
      const float mnew = fmaxf(m[r], v);
      const float sc = __expf(m[r] - mnew);
      const float pv = __expf(s[r] - mnew);
      float rs = pv;
      rs += __shfl_xor(rs, 1, 32);
      rs += __shfl_xor(rs, 2, 32);
      rs += __shfl_xor(rs, 4, 32);
      rs += __shfl_xor(rs, 8, 32);
      lsum[r] = lsum[r] * sc + rs;
      m[r] = mnew;
      scale[r] = sc;
      Pt[wv][r + 8 * hi][lrow] = pv;  // C-layout -> LDS row-major
    }
#pragma unroll
    for (int j = 0; j < 4; ++j)
#pragma unroll
      for (int r = 0; r < 8; ++r) o[j][r] *= scale[r];

    // P back as A-fragments (intra-wave LDS ops are in-order)
    v2f pa[4];
#pragma unroll
    for (int kc = 0; kc < 4; ++kc)
      pa[kc] = *(const v2f*)&Pt[wv][lrow][kc * 4 + 2 * hi];

    // ---- O += P @ V_tile : 4 independent chains interleaved ----
#pragma unroll
    for (int kc = 0; kc < 4; ++kc) {
      const int vr = kc * 4 + 2 * hi;
#pragma unroll
      for (int j = 0; j < 4; ++j) {
        v2f vb;
        vb.x = sV[cur][vr + 0][j * 16 + lrow];
        vb.y = sV[cur][vr + 1][j * 16 + lrow];
        o[j] = wmma_f32_4(pa[kc], vb, o[j]);
      }
    }
  }
#pragma unroll
  for (int j = 0; j < 4; ++j)
#pragma unroll
    for (int r = 0; r < 8; ++r)
      out[(bh * LSEQ + q0 + r + 8 * hi) * DH + j * 16 + lrow] =
          o[j][r] / lsum[r];
}

// ============================================================
extern "C" void kernel_launch(void* const* d_in, const int* in_sizes, int n_in,
                              void* d_out, int out_size, void* d_ws,
                              size_t ws_size, hipStream_t stream) {
  const float* Q = (const float*)d_in[0];
  const float* K = (const float*)d_in[1];
  const float* V = (const float*)d_in[2];
  const float* w_pk = (const float*)d_in[3];
  const float* b_pk = (const float*)d_in[4];
  const float* w_pb = (const float*)d_in[5];
  const float* b_pb = (const float*)d_in[6];
  const float* w_ck = (const float*)d_in[7];
  const float* b_ckv = (const float*)d_in[8];
  const float* w_cq = (const float*)d_in[9];
  const float* b_cqv = (const float*)d_in[10];
  float* out = (float*)d_out;
  float* ws = (float*)d_ws;

  float* acc = ws;               // 2 floats (lp, ce)
  float* mu_ws = ws + 64;        // 8*1024*5 = 40960
  float* q2_ws = mu_ws + 40960;  // 8*5*1024*5 = 204800
  float* CC2 = q2_ws + 204800;   // 8*8*1024*64 = 4194304

  kInit<<<1, 2, 0, stream>>>(acc);
  kA_clusters<<<NB * LSEQ, 32, 0, stream>>>(K, w_pk, b_pk, w_ck, b_ckv, w_cq,
                                            b_cqv, mu_ws, q2_ws, acc);
  kB_ce<<<NB * NCL, 256, 0, stream>>>(mu_ws, acc);
  kC_center<<<(NB * NH * LSEQ * DH) / 256, 256, 0, stream>>>(q2_ws, w_pb, b_pb,
                                                             CC2);
  kE_loss<<<1, 1, 0, stream>>>(acc, out);
  kD_attn<<<(NB * NH * 64) / 8, 256, 0, stream>>>(Q, V, CC2, out);
}